// BitPredictor_18975165514433
// MI455X (gfx1250) — compile-verified
//
#include <hip/hip_runtime.h>
#include <math.h>

// ---------------------------------------------------------------------------
// Scalar LSTM recurrence, 8192 strictly-sequential steps (serial chain).
// One wave, lane 0 only. Strategy:
//  * state (c, h) pinned to VGPRs -> whole chain stays on VALU, no per-step
//    VALU<->SALU readfirstlane hazards; loop control co-executes on SALU.
//  * every activation is exactly one gfx1250 V_TANH_F32 trans op;
//    sigmoid input scale pre-folded:  sigmoid(h*w+b)=0.5*tanh(h*w/2+b/2)+0.5
//  * manual unroll x4 + one global_store_b128 per 4 steps; outer loop kept
//    tight (#pragma unroll 1) since a serial chain gains zero ILP from
//    further unrolling, and trip-count assumes kill loop versioning.
// ---------------------------------------------------------------------------

__device__ __forceinline__ float fast_tanh(float x) {
#if __has_builtin(__builtin_amdgcn_tanhf)
    return __builtin_amdgcn_tanhf(x);
#elif __has_builtin(__builtin_amdgcn_tanh_f32)
    return __builtin_amdgcn_tanh_f32(x);
#else
    return tanhf(x);   // ocml fallback if no hw tanh builtin
#endif
}

// Pin a value into a VGPR; marks it divergent so the dependent recurrence
// stays on the VALU instead of being scalarized onto the SALU float pipe.
__device__ __forceinline__ float vgpr_zero() {
    float y;
    asm("v_mov_b32 %0, 0" : "=v"(y));
    return y;
}

__global__ void __launch_bounds__(32, 1)
lstm_bitpred_seq_kernel(const float* __restrict__ Wi,
                        const float* __restrict__ Wh,
                        const float* __restrict__ b,
                        float* __restrict__ out,
                        int features) {
    if (threadIdx.x != 0 || blockIdx.x != 0) return;

    // Launch contract: features == out_size == 8192 (positive, multiple of 4).
    __builtin_assume(features >= 4);
    __builtin_assume((features & 3) == 0);

    // Identity: fed-back output is BOTH x and h every step (both init 0), so
    // z = x*Wi + h*Wh + b == h*(Wi+Wh) + b.  Fold weights once; pre-halve the
    // sigmoid gates (i, f, o) so each activation is a bare tanh + one affine.
    const float wi_h = 0.5f * (Wi[0] + Wh[0]);   // input gate   (sigmoid)
    const float wf_h = 0.5f * (Wi[1] + Wh[1]);   // forget gate  (sigmoid)
    const float wg   =         Wi[2] + Wh[2];    // cell gate    (tanh)
    const float wo_h = 0.5f * (Wi[3] + Wh[3]);   // output gate  (sigmoid)
    const float bi_h = 0.5f * b[0];
    const float bf_h = 0.5f * b[1];
    const float bg   =        b[2];
    const float bo_h = 0.5f * b[3];

    float c = vgpr_zero();   // VGPR-resident state -> pure-VALU chain
    float h = vgpr_zero();

    const int n4 = features >> 2;
    __builtin_assume(n4 >= 1);
    float4* __restrict__ out4 = reinterpret_cast<float4*>(out);

#pragma unroll 1
    for (int q = 0; q < n4; ++q) {
        float4 r;
#pragma unroll
        for (int u = 0; u < 4; ++u) {
            // Gate pre-activations: 4 independent v_fmac off VGPR-resident h
            // with SGPR-resident weights/biases as direct VALU sources.
            const float zi = fmaf(h, wi_h, bi_h);
            const float zf = fmaf(h, wf_h, bf_h);
            const float zg = fmaf(h, wg,   bg);
            const float zo = fmaf(h, wo_h, bo_h);
            // 4 independent v_tanh_f32 issue back-to-back; latencies overlap.
            const float ti = fast_tanh(zi);
            const float tf = fast_tanh(zf);
            const float g  = fast_tanh(zg);
            const float to = fast_tanh(zo);
            const float ig = fmaf(0.5f, ti, 0.5f);   // sigmoid affine
            const float fg = fmaf(0.5f, tf, 0.5f);
            const float og = fmaf(0.5f, to, 0.5f);
            c = fmaf(fg, c, ig * g);
            h = og * fast_tanh(c);
            (&r.x)[u] = h;
        }
        out4[q] = r;                          // one b128 store per 4 steps
    }
}

extern "C" void kernel_launch(void* const* d_in, const int* in_sizes, int n_in,
                              void* d_out, int out_size, void* d_ws, size_t ws_size,
                              hipStream_t stream) {
    (void)in_sizes; (void)n_in; (void)d_ws; (void)ws_size;
    const float* Wi = (const float*)d_in[0];  // (1,4)
    const float* Wh = (const float*)d_in[1];  // (1,4)
    const float* b  = (const float*)d_in[2];  // (4,)
    float* out = (float*)d_out;               // (features,)
    // features == out_size (8192), known host-side — avoids a device-side read.
    lstm_bitpred_seq_kernel<<<1, 32, 0, stream>>>(Wi, Wh, b, out, out_size);
}